// Quantizer_29910152249563
// MI455X (gfx1250) — compile-verified
//
#include <hip/hip_runtime.h>

// Quantizer for MI455X (gfx1250, wave32).
// inputs (1024*512 f32), codebook (256 f32) ->
//   d_out = [soft (N*256) | hard (N*256) | quantized (N)], N = 524288.
// Store-bandwidth bound (~1.08 GB writes @ 23.3 TB/s ~ 46us floor):
// single pass, coalesced 128B/wave accesses, non-temporal stores.
// quantized = soft x codebook via V_WMMA_F32_16X16X4_F32.

#define KDIM 256
#define TEMP 1.0e8f
#define ROWS_PER_BLOCK 16     // 16 waves per block, one element per wave
#define LDS_STRIDE 260        // 256 + pad: spreads rows across LDS banks

typedef __attribute__((ext_vector_type(2))) float v2f;
typedef __attribute__((ext_vector_type(8))) float v8f;

__global__ __launch_bounds__(512, 1) void quantizer_kernel(
    const float* __restrict__ inputs,
    const float* __restrict__ codebook,
    float* __restrict__ soft,   // N*K
    float* __restrict__ hard,   // N*K
    float* __restrict__ quant,  // N
    int N)
{
  __shared__ float soft_lds[ROWS_PER_BLOCK * LDS_STRIDE];
  __shared__ float cb_lds[KDIM];

  const int lane = threadIdx.x & 31;
  const int wave = threadIdx.x >> 5;                    // 0..15 = row in block
  int row = blockIdx.x * ROWS_PER_BLOCK + wave;
  if (row >= N) row = N - 1;                            // grid is exact; safety only

  // Stage codebook to LDS (for WMMA B-fragments) and registers (per-lane k's).
  if (threadIdx.x < KDIM) cb_lds[threadIdx.x] = codebook[threadIdx.x];

  float cbr[8];
#pragma unroll
  for (int j = 0; j < 8; ++j) cbr[j] = codebook[lane + 32 * j];  // coalesced

  const float x = inputs[row];                          // wave-broadcast load

  // distances d[k] = exp(-|x - cb[k]|), k = lane + 32*j
  float d[8];
  float m = -1.0f;
  int kmax = 0x7fffffff;
#pragma unroll
  for (int j = 0; j < 8; ++j) {
    d[j] = __expf(-fabsf(x - cbr[j]));
    const int k = lane + 32 * j;
    if (d[j] > m) { m = d[j]; kmax = k; }               // j ascending => lowest k kept
  }
  // wave32 max-reduce with lowest-index tie break (matches jnp.argmax)
#pragma unroll
  for (int off = 16; off > 0; off >>= 1) {
    const float om = __shfl_xor(m, off, 32);
    const int   ok = __shfl_xor(kmax, off, 32);
    if (om > m || (om == m && ok < kmax)) { m = om; kmax = ok; }
  }

  // softmax(TEMP * d) with max subtraction
  float s = 0.0f;
#pragma unroll
  for (int j = 0; j < 8; ++j) {
    d[j] = __expf(TEMP * (d[j] - m));
    s += d[j];
  }
#pragma unroll
  for (int off = 16; off > 0; off >>= 1) s += __shfl_xor(s, off, 32);
  const float inv = 1.0f / s;

  // streaming writes: soft + one-hot hard (contiguous 128B per wave per j)
  float* __restrict__ srow = soft + (size_t)row * KDIM;
  float* __restrict__ hrow = hard + (size_t)row * KDIM;
#pragma unroll
  for (int j = 0; j < 8; ++j) {
    const int k = lane + 32 * j;
    const float sv = d[j] * inv;
    __builtin_nontemporal_store(sv, srow + k);
    __builtin_nontemporal_store((k == kmax) ? 1.0f : 0.0f, hrow + k);
    soft_lds[wave * LDS_STRIDE + k] = sv;               // bank-conflict-free
  }

  __syncthreads();

  // quantized = soft(16xK) x cb(Kx1), via 64 chained V_WMMA_F32_16X16X4_F32.
  // Executed by all waves (EXEC must be all 1s for WMMA); redundant work is
  // trivial (2 KFLOP/wmma). Only wave 0 commits results.
  //
  // A (16x4 f32): lanes 0-15 = rows, VGPR0/1 = K 4s,4s+1; lanes 16-31 = K 4s+2,4s+3.
  // B (4x16 f32): column-broadcast codebook chunk, same half-wave split.
  const int half = lane >> 4;      // 0: lanes 0-15, 1: lanes 16-31
  const int r    = lane & 15;      // row index for A
  v8f acc = {};
#pragma unroll 8
  for (int step = 0; step < 64; ++step) {
    const int kb = 4 * step + 2 * half;
    v2f a, b;
    a.x = soft_lds[r * LDS_STRIDE + kb];
    a.y = soft_lds[r * LDS_STRIDE + kb + 1];
    b.x = cb_lds[kb];                                   // broadcast within half
    b.y = cb_lds[kb + 1];
    acc = __builtin_amdgcn_wmma_f32_16x16x4_f32(
        /*neg_a=*/false, a, /*neg_b=*/false, b,
        /*c_mod=*/(short)0, acc, /*reuse_a=*/false, /*reuse_b=*/false);
  }

  // D layout: VGPR v, lanes 0-15 -> M=v; lanes 16-31 -> M=v+8 (all N cols equal).
  if (wave == 0) {
    const int base = blockIdx.x * ROWS_PER_BLOCK;
    if (lane == 0) {
#pragma unroll
      for (int v = 0; v < 8; ++v)
        __builtin_nontemporal_store(acc[v], quant + base + v);
    } else if (lane == 16) {
#pragma unroll
      for (int v = 0; v < 8; ++v)
        __builtin_nontemporal_store(acc[v], quant + base + 8 + v);
    }
  }
}

extern "C" void kernel_launch(void* const* d_in, const int* in_sizes, int n_in,
                              void* d_out, int out_size, void* d_ws, size_t ws_size,
                              hipStream_t stream) {
  const float* inputs   = (const float*)d_in[0];   // (1024, 512)
  const float* codebook = (const float*)d_in[1];   // (1, 256)
  const int N = in_sizes[0];                       // 524288 elements

  float* soft  = (float*)d_out;                    // N*256
  float* hard  = soft + (size_t)N * KDIM;          // N*256
  float* quant = hard + (size_t)N * KDIM;          // N

  const int blocks = (N + ROWS_PER_BLOCK - 1) / ROWS_PER_BLOCK;  // 32768
  quantizer_kernel<<<dim3(blocks), dim3(512), 0, stream>>>(
      inputs, codebook, soft, hard, quant, N);
}